// VGAELinkPredictor_65000035058078
// MI455X (gfx1250) — compile-verified
//
#include <hip/hip_runtime.h>
#include <hip/hip_bf16.h>
#include <math.h>

typedef float v2f __attribute__((ext_vector_type(2)));
typedef float v8f __attribute__((ext_vector_type(8)));

#define IN_CH 128
#define OUT_CH 32

// ---------------- degree / normalization ----------------

__global__ void k_init_deg(float* __restrict__ deg, int n) {
  int i = blockIdx.x * blockDim.x + threadIdx.x;
  if (i < n) deg[i] = 1.0f;  // self-loop contributes 1
}

__global__ void k_count_deg(const int* __restrict__ dst, float* __restrict__ deg, int E) {
  int i = blockIdx.x * blockDim.x + threadIdx.x;
  if (i < E) atomicAdd(&deg[dst[i]], 1.0f);
}

__global__ void k_dinv(float* __restrict__ deg, int n) {
  int i = blockIdx.x * blockDim.x + threadIdx.x;
  if (i < n) deg[i] = rsqrtf(deg[i]);  // deg >= 1 always (self loop)
}

// ---------------- h = x @ W via fp32 WMMA 16x16x4 ----------------
// One wave computes one 16x16 tile of h. 100000/16 = 6250 row tiles,
// 32/16 = 2 col tiles -> 12500 waves total. K=128 unrolled as 32 WMMAs.

__global__ void __launch_bounds__(256)
k_gemm_xw(const float* __restrict__ x, const float* __restrict__ W,
          float* __restrict__ h, int nTiles) {
  const int wave = (int)((blockIdx.x * blockDim.x + threadIdx.x) >> 5);  // wave-uniform
  const int lane = (int)(threadIdx.x & 31);
  if (wave >= nTiles) return;  // uniform per-wave: EXEC stays all-ones

  const int rowTile = wave >> 1;
  const int colTile = wave & 1;
  const int row0 = rowTile << 4;
  const int n0 = colTile << 4;

  const int m = lane & 15;          // row within tile (A), col within tile (B/C)
  const int kHalf = (lane >> 4) << 1;  // 0 for lanes 0-15, 2 for lanes 16-31

  const float* __restrict__ xr = x + (size_t)(row0 + m) * IN_CH;
  const float* __restrict__ wc = W + n0 + m;  // column n0+m of W (row-major K x N)

  v8f c = {};
#pragma unroll
  for (int k = 0; k < IN_CH; k += 4) {
    v2f a, b;
    // A 16x4 fp32 fragment (ISA layout): lane holds K = k+kHalf, k+kHalf+1 of row m
    a.x = xr[k + kHalf];
    a.y = xr[k + kHalf + 1];
    // B 4x16 fp32 fragment: lane holds rows K = k+kHalf, k+kHalf+1 of column n0+m
    b.x = wc[(size_t)(k + kHalf) * OUT_CH];
    b.y = wc[(size_t)(k + kHalf + 1) * OUT_CH];
    c = __builtin_amdgcn_wmma_f32_16x16x4_f32(false, a, false, b,
                                              (short)0, c, false, false);
  }

  // C/D layout: VGPR r -> M = r (lanes 0-15) or r+8 (lanes 16-31), N = lane%16
  const int nCol = n0 + m;
  const int mBase = row0 + ((lane >> 4) << 3);
#pragma unroll
  for (int r = 0; r < 8; ++r) {
    h[(size_t)(mBase + r) * OUT_CH + nCol] = c[r];
  }
}

// ---------------- self-loop init: z = h * dinv^2 ----------------

__global__ void k_selfloop(const float* __restrict__ h, const float* __restrict__ dinv,
                           float* __restrict__ z, int total) {
  int i = blockIdx.x * blockDim.x + threadIdx.x;
  if (i < total) {
    int node = i >> 5;
    float dv = dinv[node];
    z[i] = h[i] * dv * dv;
  }
}

// ---------------- edge scatter: z[dst] += h[src] * dinv[src]*dinv[dst] ----------------
// One wave32 per edge: 32 lanes = 32 channels -> 128B coalesced gather + 32 atomics
// into the L2-resident z array.

__global__ void __launch_bounds__(256)
k_scatter(const int* __restrict__ src, const int* __restrict__ dst,
          const float* __restrict__ h, const float* __restrict__ dinv,
          float* __restrict__ z, int E) {
  const int e = (int)((blockIdx.x * blockDim.x + threadIdx.x) >> 5);
  const int lane = (int)(threadIdx.x & 31);
  if (e >= E) return;
  const int s = src[e];
  const int d = dst[e];
  const float norm = dinv[s] * dinv[d];
  const float v = h[(size_t)s * OUT_CH + lane] * norm;
  atomicAdd(&z[(size_t)d * OUT_CH + lane], v);
}

// ---------------- bias + ReLU ----------------

__global__ void k_bias_relu(float* __restrict__ z, const float* __restrict__ b, int total) {
  int i = blockIdx.x * blockDim.x + threadIdx.x;
  if (i < total) {
    float v = z[i] + b[i & (OUT_CH - 1)];
    z[i] = v > 0.0f ? v : 0.0f;
  }
}

// ---------------- decoder: sigmoid(<z[src], z[dst]>) ----------------
// One wave32 per edge, shuffle-tree reduction over the 32 channels.

__global__ void __launch_bounds__(256)
k_decode(const int* __restrict__ src, const int* __restrict__ dst,
         const float* __restrict__ z, float* __restrict__ out, int E) {
  const int e = (int)((blockIdx.x * blockDim.x + threadIdx.x) >> 5);
  const int lane = (int)(threadIdx.x & 31);
  if (e >= E) return;
  const int s = src[e];
  const int d = dst[e];
  float p = z[(size_t)s * OUT_CH + lane] * z[(size_t)d * OUT_CH + lane];
#pragma unroll
  for (int off = 16; off > 0; off >>= 1) {
    p += __shfl_xor(p, off, 32);
  }
  if (lane == 0) {
    out[e] = 1.0f / (1.0f + __expf(-p));
  }
}

// ---------------- launcher ----------------

extern "C" void kernel_launch(void* const* d_in, const int* in_sizes, int n_in,
                              void* d_out, int out_size, void* d_ws, size_t ws_size,
                              hipStream_t stream) {
  const float* x = (const float*)d_in[0];
  const int* ei = (const int*)d_in[1];   // [2, E] int32
  const float* W = (const float*)d_in[2];
  const float* b = (const float*)d_in[3];
  float* out = (float*)d_out;

  const int N = in_sizes[0] / IN_CH;   // 100000
  const int E = in_sizes[1] / 2;       // 1600000
  const int* src = ei;
  const int* dst = ei + E;

  // workspace layout
  char* ws = (char*)d_ws;
  float* dinv = (float*)ws;                             // N floats (used as deg then dinv)
  float* h = (float*)(ws + (1u << 19));                 // N*32 floats (12.8 MB)
  float* z = (float*)(ws + (1u << 19) + (size_t)N * OUT_CH * sizeof(float));

  const int B = 256;

  // degrees -> dinv
  k_init_deg<<<(N + B - 1) / B, B, 0, stream>>>(dinv, N);
  k_count_deg<<<(E + B - 1) / B, B, 0, stream>>>(dst, dinv, E);
  k_dinv<<<(N + B - 1) / B, B, 0, stream>>>(dinv, N);

  // h = x @ W  (WMMA fp32)
  const int nTiles = (N / 16) * (OUT_CH / 16);  // 12500 waves
  const int wavesPerBlock = B / 32;
  k_gemm_xw<<<(nTiles + wavesPerBlock - 1) / wavesPerBlock, B, 0, stream>>>(x, W, h, nTiles);

  // aggregate
  const int total = N * OUT_CH;
  k_selfloop<<<(total + B - 1) / B, B, 0, stream>>>(h, dinv, z, total);
  k_scatter<<<((size_t)E * 32 + B - 1) / B, B, 0, stream>>>(src, dst, h, dinv, z, E);
  k_bias_relu<<<(total + B - 1) / B, B, 0, stream>>>(z, b, total);

  // decode
  k_decode<<<((size_t)E * 32 + B - 1) / B, B, 0, stream>>>(src, dst, z, out, E);
}